// model2_45569603010745
// MI455X (gfx1250) — compile-verified
//
#include <hip/hip_runtime.h>

typedef float v2f __attribute__((ext_vector_type(2)));
typedef float v8f __attribute__((ext_vector_type(8)));

namespace {
constexpr int kB   = 8;
constexpr int kGXY = 22;
constexpr int kV   = kGXY * kGXY;   // 484 voxels per graph
constexpr int kS   = kB * kV;       // 3872 pooled "nodes"
constexpr float kSX = 16.0f / 346.0f;
constexpr float kSY = 12.0f / 260.0f;
}

// ---- order-preserving float<->uint encoding for atomic max --------------
__device__ __forceinline__ unsigned f2ord(float f) {
  unsigned b = __float_as_uint(f);
  return (b & 0x80000000u) ? ~b : (b | 0x80000000u);
}
__device__ __forceinline__ float ord2f(unsigned u) {
  return (u & 0x80000000u) ? __uint_as_float(u & 0x7fffffffu) : __uint_as_float(~u);
}

// ---- generic utilities ---------------------------------------------------
__global__ void k_zero(unsigned* __restrict__ p, long long n) {
  long long i = (long long)blockIdx.x * blockDim.x + threadIdx.x;
  if (i < n) p[i] = 0u;
}
__global__ void k_setf(float* p, float v) {
  if (threadIdx.x == 0 && blockIdx.x == 0) *p = v;
}
__global__ void k_add(float* __restrict__ a, const float* __restrict__ b, int n) {
  int i = blockIdx.x * blockDim.x + threadIdx.x;
  if (i < n) a[i] += b[i];
}

static inline void zero_dev(void* p, size_t words, hipStream_t s) {
  k_zero<<<dim3((unsigned)((words + 255) / 256)), 256, 0, s>>>((unsigned*)p,
                                                               (long long)words);
}

// ---- in-degree (layer-invariant) ----------------------------------------
__global__ void k_deg(const int* __restrict__ dst, float* __restrict__ deg, int E) {
  int e = blockIdx.x * blockDim.x + threadIdx.x;
  if (e < E) atomicAdd(&deg[dst[e]], 1.0f);
}

// ---- node transform: Y[M,4*Cout] = X[M,K] @ Wflat, via f32 WMMA ---------
// W stored [4, K, Cout]; flat column j = kk*Cout + co -> W[(kk*K + k)*Cout + co].
// Templated on (K, Cout): compile-time-unrolled K loop, register-resident B
// fragments reused across a strip of M tiles, A loaded as one b64 per step.
template <int K, int Cout>
__global__ void k_gemm(const float* __restrict__ X, const float* __restrict__ W,
                       float* __restrict__ Y, int Mtiles) {
  constexpr int Nc = 4 * Cout;
  constexpr int KT = (K + 3) / 4;
  const int lane = threadIdx.x;
  const int m    = lane & 15;
  const int hi   = lane >> 4;
  const int col0 = blockIdx.y * 16;
  const int j    = col0 + m;           // global output column
  const int kk   = j / Cout;
  const int co   = j % Cout;

  // Preload B fragments (weights) once per wave; reused for every M tile.
  v2f bf[KT];
#pragma unroll
  for (int t = 0; t < KT; ++t) {
    const int k0 = 4 * t + 2 * hi;
    const int k1 = k0 + 1;
    v2f b;
    b.x = (k0 < K) ? W[(size_t)(kk * K + k0) * Cout + co] : 0.0f;
    b.y = (k1 < K) ? W[(size_t)(kk * K + k1) * Cout + co] : 0.0f;
    bf[t] = b;
  }

  for (int tm = blockIdx.x; tm < Mtiles; tm += gridDim.x) {
    const int row0 = tm * 16;
    const float* xr = X + (size_t)(row0 + m) * K;
    v8f c = {};
#pragma unroll
    for (int t = 0; t < KT; ++t) {
      v2f a;
      if (K == 1) {                    // compile-time branch
        a.x = (hi == 0) ? xr[0] : 0.0f;
        a.y = 0.0f;
      } else {
        a = *(const v2f*)(xr + 4 * t + 2 * hi);   // contiguous, 8B aligned
      }
      c = __builtin_amdgcn_wmma_f32_16x16x4_f32(false, a, false, bf[t],
                                                (short)0, c, false, false);
    }
    float* yr = Y + (size_t)row0 * Nc + col0 + m;
#pragma unroll
    for (int r = 0; r < 8; ++r)
      yr[(size_t)(r + 8 * hi) * Nc] = c[r];
  }
}

// ---- edge message + scatter: one thread per (edge, channel) -------------
__global__ void k_edge(const float* __restrict__ Y, const float* __restrict__ ea,
                       const int* __restrict__ src, const int* __restrict__ dst,
                       float* __restrict__ acc, int total, int E, int Cout,
                       int shift) {
  int gid = blockIdx.x * blockDim.x + threadIdx.x;
  if (gid >= total) return;
  int ch = gid & (Cout - 1);
  int e  = gid >> shift;
  // Hide L2 gather latency: prefetch a future edge's source row.
  int ep = e + 64;
  if (ch == 0 && ep < E)
    __builtin_prefetch(Y + (size_t)src[ep] * (Cout << 2), 0, 1);
  float u = ea[2 * e], v = ea[2 * e + 1];
  float w0 = (1.0f - u) * (1.0f - v);
  float w1 = u * (1.0f - v);
  float w2 = (1.0f - u) * v;
  float w3 = u * v;
  const float* yb = Y + (size_t)src[e] * (Cout << 2);
  float msg = w0 * yb[ch] + w1 * yb[Cout + ch] + w2 * yb[2 * Cout + ch] +
              w3 * yb[3 * Cout + ch];
  atomicAdd(&acc[(size_t)dst[e] * Cout + ch], msg);
}

// ---- mean-normalize + ELU + BN partial sums (LDS-reduced) ---------------
__global__ void k_fin_stats(const float* __restrict__ acc, const float* __restrict__ deg,
                            float* __restrict__ h, float* __restrict__ st,
                            int nelem, int C) {
  __shared__ float s[64];
  int t = threadIdx.x;
  if (t < 64) s[t] = 0.0f;
  __syncthreads();
  int i = blockIdx.x * blockDim.x + t;
  if (i < nelem) {
    int ch   = i & (C - 1);
    int node = i / C;
    float d = deg[node];
    if (d < 1.0f) d = 1.0f;
    float v = acc[i] / d;
    v = (v > 0.0f) ? v : (expf(v) - 1.0f);   // ELU
    h[i] = v;
    atomicAdd(&s[ch], v);
    atomicAdd(&s[32 + ch], v * v);
  }
  __syncthreads();
  if (t < 64) atomicAdd(&st[t], s[t]);
}

// st layout: [0..32) sum, [32..64) sumsq, [64..96) scale, [96..128) shift, [128] count
__global__ void k_bn_params(float* __restrict__ st, const float* __restrict__ g,
                            const float* __restrict__ b, int C) {
  int c = threadIdx.x;
  if (c >= C) return;
  float n    = st[128];
  float mean = st[c] / n;
  float var  = st[32 + c] / n - mean * mean;
  float sc   = rsqrtf(var + 1e-5f) * g[c];
  st[64 + c] = sc;
  st[96 + c] = b[c] - mean * sc;
}

__global__ void k_bn_apply(float* __restrict__ h, const float* __restrict__ st,
                           int nelem, int C) {
  int i = blockIdx.x * blockDim.x + threadIdx.x;
  if (i >= nelem) return;
  int ch = i & (C - 1);
  h[i] = h[i] * st[64 + ch] + st[96 + ch];
}

// ---- pool5: voxel-grid max pool ------------------------------------------
__global__ void k_pool_node(const float* __restrict__ h, const float* __restrict__ pos,
                            const int* __restrict__ batch, float* __restrict__ cnt,
                            float* __restrict__ posSum, unsigned* __restrict__ xPenc,
                            int* __restrict__ cnode, int n) {
  int gid = blockIdx.x * blockDim.x + threadIdx.x;
  if (gid >= n * 32) return;
  int ch = gid & 31, node = gid >> 5;
  float p0 = pos[2 * node], p1 = pos[2 * node + 1];
  int cx = (int)(p0 / kSX); cx = cx < 0 ? 0 : (cx > kGXY - 1 ? kGXY - 1 : cx);
  int cy = (int)(p1 / kSY); cy = cy < 0 ? 0 : (cy > kGXY - 1 ? kGXY - 1 : cy);
  int c = cx * kGXY + cy;
  int g = batch[node] * kV + c;
  if (ch == 0) {
    cnode[node] = c;
    atomicAdd(&cnt[g], 1.0f);
    atomicAdd(&posSum[2 * g], p0);
    atomicAdd(&posSum[2 * g + 1], p1);
  }
  atomicMax(&xPenc[(size_t)g * 32 + ch], f2ord(h[(size_t)node * 32 + ch]));
}

__global__ void k_pool_fin(const float* __restrict__ cnt, const float* __restrict__ posSum,
                           const unsigned* __restrict__ xPenc, float* __restrict__ posP,
                           int* __restrict__ occ, float* __restrict__ xP) {
  int gid = blockIdx.x * blockDim.x + threadIdx.x;
  if (gid >= kS * 32) return;
  int ch = gid & 31, g = gid >> 5;
  float c = cnt[g];
  int o = c > 0.0f;
  if (ch == 0) {
    float d = c < 1.0f ? 1.0f : c;
    posP[2 * g]     = posSum[2 * g] / d;
    posP[2 * g + 1] = posSum[2 * g + 1] / d;
    occ[g] = o;
  }
  xP[(size_t)g * 32 + ch] = o ? ord2f(xPenc[(size_t)g * 32 + ch]) : 0.0f;
}

// ---- coalesced pooled adjacency (dedup via plain stores, no self loops) --
__global__ void k_adj(const int* __restrict__ src, const int* __restrict__ dst,
                      const int* __restrict__ batch, const int* __restrict__ cnode,
                      unsigned char* __restrict__ adj, int E) {
  int e = blockIdx.x * blockDim.x + threadIdx.x;
  if (e >= E) return;
  int s = src[e], d = dst[e];
  int ci = cnode[d], cj = cnode[s];
  if (ci == cj) return;
  adj[(size_t)batch[s] * kV * kV + (size_t)ci * kV + cj] = 1;
}

// ---- global max |cart| over adjacency (for Cartesian norm) ---------------
__global__ void k_mx(const unsigned char* __restrict__ adj, const float* __restrict__ posP,
                     unsigned* __restrict__ mxEnc) {
  __shared__ unsigned sm;
  if (threadIdx.x == 0) sm = 0u;
  __syncthreads();
  int gid = blockIdx.x * blockDim.x + threadIdx.x;
  if (gid < kB * kV * kV && adj[gid]) {
    int j = gid % kV;
    int t = gid / kV;
    int i = t % kV;
    int b = t / kV;
    const float* pj = posP + 2 * (b * kV + j);
    const float* pi = posP + 2 * (b * kV + i);
    float dx = fabsf(pj[0] - pi[0]);
    float dy = fabsf(pj[1] - pi[1]);
    atomicMax(&sm, f2ord(dx > dy ? dx : dy));
  }
  __syncthreads();
  if (threadIdx.x == 0 && sm) atomicMax(mxEnc, sm);
}

// ---- dense pooled SplineConv + ELU + masked-BN stats ---------------------
__global__ void k_dense_conv(const float* __restrict__ yP, const unsigned char* __restrict__ adj,
                             const float* __restrict__ posP, const unsigned* __restrict__ mxEnc,
                             const int* __restrict__ occ, float* __restrict__ hOut,
                             float* __restrict__ st) {
  int bi = blockIdx.x;            // 0..kS-1
  int b  = bi / kV;
  int i  = bi % kV;
  int ch = threadIdx.x;           // 0..31
  float mx  = ord2f(*mxEnc);
  float inv = 1.0f / (2.0f * mx);
  float pix = posP[2 * bi], piy = posP[2 * bi + 1];
  float accv = 0.0f, degc = 0.0f;
  const unsigned char* arow = adj + (size_t)b * kV * kV + (size_t)i * kV;
  for (int j = 0; j < kV; ++j) {
    if (!arow[j]) continue;
    degc += 1.0f;
    int bj = b * kV + j;
    float u = (posP[2 * bj] - pix) * inv + 0.5f;
    float v = (posP[2 * bj + 1] - piy) * inv + 0.5f;
    float w0 = (1.0f - u) * (1.0f - v);
    float w1 = u * (1.0f - v);
    float w2 = (1.0f - u) * v;
    float w3 = u * v;
    const float* yb = yP + (size_t)bj * 128;
    accv += w0 * yb[ch] + w1 * yb[32 + ch] + w2 * yb[64 + ch] + w3 * yb[96 + ch];
  }
  float d = degc < 1.0f ? 1.0f : degc;
  float val = accv / d;
  val = (val > 0.0f) ? val : (expf(val) - 1.0f);   // ELU
  hOut[(size_t)bi * 32 + ch] = val;
  if (occ[bi]) {
    atomicAdd(&st[ch], val);
    atomicAdd(&st[32 + ch], val * val);
    if (ch == 0) atomicAdd(&st[128], 1.0f);
  }
}

// ---- MaxPoolingX: 4x4 grid max over occupied voxels ----------------------
__global__ void k_poolx(const float* __restrict__ h2, const float* __restrict__ posP,
                        const int* __restrict__ occ, unsigned* __restrict__ pfEnc) {
  int gid = blockIdx.x * blockDim.x + threadIdx.x;
  if (gid >= kS * 32) return;
  int ch = gid & 31, g = gid >> 5;
  if (!occ[g]) return;
  int b = g / kV;
  int fx = (int)(posP[2 * g] * 4.0f);     fx = fx < 0 ? 0 : (fx > 3 ? 3 : fx);
  int fy = (int)(posP[2 * g + 1] * 4.0f); fy = fy < 0 ? 0 : (fy > 3 ? 3 : fy);
  int c2 = fx * 4 + fy;
  atomicMax(&pfEnc[(size_t)(b * 16 + c2) * 32 + ch], f2ord(h2[(size_t)g * 32 + ch]));
}

// ---- final FC: [B,512] @ Wfc.T -> [B,10] ---------------------------------
__global__ void k_fc(const unsigned* __restrict__ pfEnc, const float* __restrict__ Wfc,
                     float* __restrict__ out) {
  int t = threadIdx.x;
  if (t >= kB * 10) return;
  int b = t / 10, o = t % 10;
  float s = 0.0f;
  for (int i = 0; i < 512; ++i) {
    float v = ord2f(pfEnc[b * 512 + i]);
    if (!(v > -INFINITY && v < INFINITY)) v = 0.0f;  // non-finite -> 0
    s += v * Wfc[o * 512 + i];
  }
  out[t] = s;
}

// ---- one sparse SplineConv + ELU + BN layer ------------------------------
template <int K, int Cout>
static void run_sparse_layer(const float* Xin, const float* W, const float* g,
                             const float* b, float* Hout, float* st,
                             const float* ea, const int* src, const int* dst,
                             float* Y, float* acc, const float* deg, int NT,
                             int E, hipStream_t stream) {
  zero_dev(acc, (size_t)NT * Cout, stream);
  k_setf<<<1, 1, 0, stream>>>(st + 128, (float)NT);
  int Mtiles = NT / 16;
  int gx = Mtiles < 2048 ? Mtiles : 2048;
  k_gemm<K, Cout><<<dim3(gx, (4 * Cout) / 16), 32, 0, stream>>>(Xin, W, Y,
                                                                Mtiles);
  constexpr int shift = (Cout == 8) ? 3 : ((Cout == 16) ? 4 : 5);
  int total = E * Cout;
  k_edge<<<(total + 255) / 256, 256, 0, stream>>>(Y, ea, src, dst, acc, total,
                                                  E, Cout, shift);
  int nelem = NT * Cout;
  k_fin_stats<<<(nelem + 255) / 256, 256, 0, stream>>>(acc, deg, Hout, st,
                                                       nelem, Cout);
  k_bn_params<<<1, 32, 0, stream>>>(st, g, b, Cout);
  k_bn_apply<<<(nelem + 255) / 256, 256, 0, stream>>>(Hout, st, nelem, Cout);
}

// =========================================================================
extern "C" void kernel_launch(void* const* d_in, const int* in_sizes, int n_in,
                              void* d_out, int out_size, void* d_ws, size_t ws_size,
                              hipStream_t stream) {
  (void)n_in; (void)out_size; (void)ws_size;
  const float* x   = (const float*)d_in[0];
  const float* ea  = (const float*)d_in[1];
  const float* pos = (const float*)d_in[2];
  const float* Wp[8]; const float* gp[8]; const float* bp[8];
  for (int i = 0; i < 8; ++i) {
    Wp[i] = (const float*)d_in[3 + 3 * i];
    gp[i] = (const float*)d_in[4 + 3 * i];
    bp[i] = (const float*)d_in[5 + 3 * i];
  }
  const float* Wfc   = (const float*)d_in[27];
  const int*   src   = (const int*)d_in[28];
  const int*   batch = (const int*)d_in[29];
  const int NT = in_sizes[0];        // 131072 (x is [NT,1])
  const int E  = in_sizes[1] / 2;    // 2097152
  const int* dst = src + E;
  float* out = (float*)d_out;

  // ---- workspace carve-out ----
  char* ws = (char*)d_ws;
  size_t off = 0;
  auto carve = [&](size_t bytes) -> char* {
    off = (off + 255) & ~(size_t)255;
    char* p = ws + off;
    off += bytes;
    return p;
  };
  float*    deg    = (float*)carve((size_t)NT * 4);
  float*    Y      = (float*)carve((size_t)NT * 128 * 4);
  float*    acc    = (float*)carve((size_t)NT * 32 * 4);
  float*    hA     = (float*)carve((size_t)NT * 32 * 4);
  float*    hB     = (float*)carve((size_t)NT * 32 * 4);
  float*    hC     = (float*)carve((size_t)NT * 32 * 4);
  float*    stats  = (float*)carve(8 * 256 * 4);
  float*    cnt    = (float*)carve((size_t)kS * 4);
  float*    posSum = (float*)carve((size_t)kS * 8);
  float*    posP   = (float*)carve((size_t)kS * 8);
  int*      occ    = (int*)carve((size_t)kS * 4);
  int*      cnode  = (int*)carve((size_t)NT * 4);
  unsigned* xPenc  = (unsigned*)carve((size_t)kS * 32 * 4);
  float*    xP     = (float*)carve((size_t)kS * 32 * 4);
  unsigned char* adj = (unsigned char*)carve((size_t)kB * kV * kV);
  unsigned* mxEnc  = (unsigned*)carve(256);
  float*    yP     = (float*)carve((size_t)kS * 128 * 4);
  float*    h2a    = (float*)carve((size_t)kS * 32 * 4);
  float*    h2b    = (float*)carve((size_t)kS * 32 * 4);
  unsigned* pfEnc  = (unsigned*)carve((size_t)kB * 16 * 32 * 4);

  // ---- per-launch deterministic re-init ----
  zero_dev(deg, NT, stream);
  zero_dev(stats, 8 * 256, stream);
  zero_dev(cnt, kS, stream);
  zero_dev(posSum, (size_t)kS * 2, stream);
  zero_dev(xPenc, (size_t)kS * 32, stream);
  zero_dev(adj, (size_t)kB * kV * kV / 4, stream);
  zero_dev(mxEnc, 1, stream);
  zero_dev(pfEnc, (size_t)kB * 16 * 32, stream);

  k_deg<<<(E + 255) / 256, 256, 0, stream>>>(dst, deg, E);

  // ---- 6 sparse SplineConv layers ----
  run_sparse_layer<1, 8>(x, Wp[0], gp[0], bp[0], hA, stats + 0 * 256, ea, src,
                         dst, Y, acc, deg, NT, E, stream);
  run_sparse_layer<8, 16>(hA, Wp[1], gp[1], bp[1], hB, stats + 1 * 256, ea, src,
                          dst, Y, acc, deg, NT, E, stream);
  run_sparse_layer<16, 16>(hB, Wp[2], gp[2], bp[2], hA, stats + 2 * 256, ea,
                           src, dst, Y, acc, deg, NT, E, stream);  // skip src
  run_sparse_layer<16, 16>(hA, Wp[3], gp[3], bp[3], hB, stats + 3 * 256, ea,
                           src, dst, Y, acc, deg, NT, E, stream);
  run_sparse_layer<16, 16>(hB, Wp[4], gp[4], bp[4], hC, stats + 4 * 256, ea,
                           src, dst, Y, acc, deg, NT, E, stream);
  k_add<<<(NT * 16 + 255) / 256, 256, 0, stream>>>(hC, hA, NT * 16);  // + skip
  run_sparse_layer<16, 32>(hC, Wp[5], gp[5], bp[5], hB, stats + 5 * 256, ea,
                           src, dst, Y, acc, deg, NT, E, stream);

  // ---- pool5 ----
  k_pool_node<<<(NT * 32 + 255) / 256, 256, 0, stream>>>(hB, pos, batch, cnt,
                                                         posSum, xPenc, cnode, NT);
  k_pool_fin<<<(kS * 32 + 255) / 256, 256, 0, stream>>>(cnt, posSum, xPenc,
                                                        posP, occ, xP);
  k_adj<<<(E + 255) / 256, 256, 0, stream>>>(src, dst, batch, cnode, adj, E);
  k_mx<<<(kB * kV * kV + 255) / 256, 256, 0, stream>>>(adj, posP, mxEnc);

  // ---- dense conv6 ----
  k_gemm<32, 32><<<dim3(kS / 16, 8), 32, 0, stream>>>(xP, Wp[6], yP, kS / 16);
  k_dense_conv<<<kS, 32, 0, stream>>>(yP, adj, posP, mxEnc, occ, h2a,
                                      stats + 6 * 256);
  k_bn_params<<<1, 32, 0, stream>>>(stats + 6 * 256, gp[6], bp[6], 32);
  k_bn_apply<<<(kS * 32 + 255) / 256, 256, 0, stream>>>(h2a, stats + 6 * 256,
                                                        kS * 32, 32);
  // ---- dense conv7 ----
  k_gemm<32, 32><<<dim3(kS / 16, 8), 32, 0, stream>>>(h2a, Wp[7], yP, kS / 16);
  k_dense_conv<<<kS, 32, 0, stream>>>(yP, adj, posP, mxEnc, occ, h2b,
                                      stats + 7 * 256);
  k_bn_params<<<1, 32, 0, stream>>>(stats + 7 * 256, gp[7], bp[7], 32);
  k_bn_apply<<<(kS * 32 + 255) / 256, 256, 0, stream>>>(h2b, stats + 7 * 256,
                                                        kS * 32, 32);
  k_add<<<(kS * 32 + 255) / 256, 256, 0, stream>>>(h2b, xP, kS * 32);  // + xP

  // ---- MaxPoolingX + FC ----
  k_poolx<<<(kS * 32 + 255) / 256, 256, 0, stream>>>(h2b, posP, occ, pfEnc);
  k_fc<<<1, 128, 0, stream>>>(pfEnc, Wfc, out);
}